// GRAPH1_77103252898361
// MI455X (gfx1250) — compile-verified
//
#include <hip/hip_runtime.h>

static constexpr int NNODE = 7936;      // 128*62, exactly 496 16-row tiles
static constexpr int FIN   = 200;
static constexpr int H     = 64;
static constexpr int NE    = 160000;
static constexpr int HEADS = 3;
static constexpr int C3    = H * HEADS; // 192

typedef float v2f __attribute__((ext_vector_type(2)));
typedef float v8f __attribute__((ext_vector_type(8)));

__device__ __forceinline__ void atomicMaxF(float* addr, float val) {
  // float max via int max (>=0) / uint min (<0); storage initialized to -inf
  if (val >= 0.0f) atomicMax((int*)addr, __float_as_int(val));
  else             atomicMin((unsigned int*)addr, __float_as_uint(val));
}
__device__ __forceinline__ float lrelu(float x) { return x > 0.0f ? x : 0.2f * x; }

// ---------------------------------------------------------------------------
// fp32 WMMA GEMM: C[496*16 x NT*16] (+)= A[M x K] * B (+ bias) (+ relu)
//   TRANSB=0: B is [K x NC] row-major;  TRANSB=1: B is [NC x K] row-major (W^T)
// One wave per 16-row stripe, NT column tiles, K stepped by 4 via
// V_WMMA_F32_16X16X4_F32. A-fragment reused across the NT column tiles.
// flags: bit0 = relu, bit1 = accumulate into C.
// ---------------------------------------------------------------------------
template<int NT, int TRANSB>
__global__ __launch_bounds__(32)
void gemm_wmma(const float* __restrict__ A, const float* __restrict__ B,
               const float* __restrict__ bias, float* __restrict__ C,
               int K, int flags)
{
  constexpr int NC = NT * 16;
  const int tileM = blockIdx.x * 16;
  const int lane  = threadIdx.x;   // 0..31, full wave, no divergence
  const int half  = lane >> 4;     // 0: K=k0,k0+1  1: K=k0+2,k0+3
  const int l     = lane & 15;
  v8f acc[NT];
#pragma unroll
  for (int t = 0; t < NT; ++t) acc[t] = (v8f){0.f,0.f,0.f,0.f,0.f,0.f,0.f,0.f};
  const float* arow = A + (size_t)(tileM + l) * K;
  for (int k0 = 0; k0 < K; k0 += 4) {
    const int ka = k0 + half * 2;
    v2f a; a.x = arow[ka]; a.y = arow[ka + 1];        // A 16x4 fragment
#pragma unroll
    for (int t = 0; t < NT; ++t) {
      const int col = t * 16 + l;
      v2f b;                                          // B 4x16 fragment
      if (TRANSB) { const float* bp = B + (size_t)col * K + ka; b.x = bp[0]; b.y = bp[1]; }
      else        { b.x = B[(size_t)ka * NC + col];  b.y = B[(size_t)(ka + 1) * NC + col]; }
      acc[t] = __builtin_amdgcn_wmma_f32_16x16x4_f32(
                 false, a, false, b, (short)0, acc[t], false, false);
    }
  }
#pragma unroll
  for (int t = 0; t < NT; ++t) {
    const int col = t * 16 + l;
    const float bv = bias ? bias[col] : 0.0f;
#pragma unroll
    for (int r = 0; r < 8; ++r) {                     // C/D: VGPR r -> M=r+half*8
      const int row = tileM + r + half * 8;
      float v = acc[t][r] + bv;
      float* cp = C + (size_t)row * NC + col;
      if (flags & 2) v += *cp;
      if (flags & 1) v = fmaxf(v, 0.0f);
      *cp = v;
    }
  }
}

// ----------------------------- elementwise ---------------------------------
__global__ void k_fill(float* p, float v, int n)
{ int i = blockIdx.x * blockDim.x + threadIdx.x; if (i < n) p[i] = v; }
__global__ void k_relu(const float* a, float* o, int n)
{ int i = blockIdx.x * blockDim.x + threadIdx.x; if (i < n) o[i] = fmaxf(a[i], 0.f); }
__global__ void k_addrelu(const float* a, const float* b, float* o, int n)
{ int i = blockIdx.x * blockDim.x + threadIdx.x; if (i < n) o[i] = fmaxf(a[i] + b[i], 0.f); }
__global__ void k_reluadd(const float* a, const float* b, float* o, int n)
{ int i = blockIdx.x * blockDim.x + threadIdx.x; if (i < n) o[i] = fmaxf(a[i], 0.f) + b[i]; }
__global__ void k_neg(const float* a, float* o, int n)
{ int i = blockIdx.x * blockDim.x + threadIdx.x; if (i < n) o[i] = -a[i]; }
__global__ void k_mulrelu(const float* a, const float* b, float* o, int n)
{ int i = blockIdx.x * blockDim.x + threadIdx.x; if (i < n) o[i] = a[i] * fmaxf(b[i], 0.f); }

// ----------------------------- graph prep ----------------------------------
__global__ void k_deg(const int* __restrict__ src, float* deg, int e)
{ int i = blockIdx.x * blockDim.x + threadIdx.x; if (i < e) atomicAdd(&deg[src[i]], 1.0f); }
__global__ void k_dis(float* d, int n)
{ int i = blockIdx.x * blockDim.x + threadIdx.x;
  if (i < n) { float v = d[i]; d[i] = v > 0.f ? rsqrtf(fmaxf(v, 1e-12f)) : 0.f; } }
__global__ void k_ew(const int* __restrict__ src, const int* __restrict__ dst,
                     const float* __restrict__ dis, float* w, int e)
{ int i = blockIdx.x * blockDim.x + threadIdx.x;
  if (i < e) w[i] = -(dis[src[i]] * dis[dst[i]]); }

// Laplacian scatter: out[dst[k],f] += scale*w[k]*z[src[k],f]   (H=64 features)
__global__ void k_lap(const int* __restrict__ src, const int* __restrict__ dst,
                      const float* __restrict__ w, const float* __restrict__ z,
                      float* out, float scale, int e)
{ int tid = blockIdx.x * blockDim.x + threadIdx.x;
  if (tid < e * H) {
    int k = tid >> 6, f = tid & 63;
    atomicAdd(&out[(size_t)dst[k] * H + f], scale * w[k] * z[(size_t)src[k] * H + f]);
  } }

// ----------------------------- batch norm ----------------------------------
__global__ void k_colstats(const float* __restrict__ X, float* s, float* q,
                           int rows, int C, int rpb)
{ int col = threadIdx.x;                     // blockDim.x == C (64 or 192)
  int r0 = blockIdx.x * rpb;
  int r1 = r0 + rpb; if (r1 > rows) r1 = rows;
  float a = 0.f, b = 0.f;
  for (int r = r0; r < r1; ++r) { float v = X[(size_t)r * C + col]; a += v; b += v * v; }
  atomicAdd(&s[col], a); atomicAdd(&q[col], b); }

__global__ void k_bnapply(float* X, const float* __restrict__ s, const float* __restrict__ q,
                          const float* __restrict__ g, const float* __restrict__ b,
                          int rows, int C)
{ int i = blockIdx.x * blockDim.x + threadIdx.x;
  if (i < rows * C) {
    int col = i % C;
    float mu  = s[col] / rows;
    float var = q[col] / rows - mu * mu;
    X[i] = g[col] * (X[i] - mu) * rsqrtf(var + 1e-5f) + b[col];
  } }

// ------------------------------- GAT ---------------------------------------
__global__ void k_att(const float* __restrict__ h, const float* __restrict__ as,
                      const float* __restrict__ ad, float* asrc, float* adst, int n)
{ int i = blockIdx.x * blockDim.x + threadIdx.x;
  if (i < n * HEADS) {
    int node = i / HEADS, hd = i % HEADS;
    const float* hp = h + (size_t)node * C3 + hd * H;
    const float* sp = as + hd * H;
    const float* dp = ad + hd * H;
    float s = 0.f, d = 0.f;
    for (int f = 0; f < H; ++f) { float v = hp[f]; s += v * sp[f]; d += v * dp[f]; }
    asrc[i] = s; adst[i] = d;
  } }

// s/d == nullptr means self-loop pass (s=d=k)
__global__ void k_gmax(const int* __restrict__ s, const int* __restrict__ d,
                       const float* __restrict__ asrc, const float* __restrict__ adst,
                       float* m, int cnt)
{ int i = blockIdx.x * blockDim.x + threadIdx.x;
  if (i < cnt * HEADS) {
    int k = i / HEADS, hd = i % HEADS;
    int ss = s ? s[k] : k, dd = d ? d[k] : k;
    atomicMaxF(&m[dd * HEADS + hd], lrelu(asrc[ss * HEADS + hd] + adst[dd * HEADS + hd]));
  } }

__global__ void k_gexp(const int* __restrict__ s, const int* __restrict__ d,
                       const float* __restrict__ asrc, const float* __restrict__ adst,
                       const float* __restrict__ m, float* p, float* denom, int cnt)
{ int i = blockIdx.x * blockDim.x + threadIdx.x;
  if (i < cnt * HEADS) {
    int k = i / HEADS, hd = i % HEADS;
    int ss = s ? s[k] : k, dd = d ? d[k] : k;
    float e  = lrelu(asrc[ss * HEADS + hd] + adst[dd * HEADS + hd]);
    float pv = __expf(e - m[dd * HEADS + hd]);
    p[i] = pv;
    atomicAdd(&denom[dd * HEADS + hd], pv);
  } }

__global__ void k_gnum(const int* __restrict__ s, const int* __restrict__ d,
                       const float* __restrict__ p, const float* __restrict__ h,
                       float* gout, int cnt)
{ int tid = blockIdx.x * blockDim.x + threadIdx.x;
  if (tid < cnt * C3) {
    int k = tid / C3, f = tid % C3, hd = f >> 6;
    int ss = s ? s[k] : k, dd = d ? d[k] : k;
    atomicAdd(&gout[(size_t)dd * C3 + f], p[k * HEADS + hd] * h[(size_t)ss * C3 + f]);
  } }

__global__ void k_gfin(float* gout, const float* __restrict__ denom,
                       const float* __restrict__ bias, int n)
{ int i = blockIdx.x * blockDim.x + threadIdx.x;
  if (i < n * C3) {
    int node = i / C3, f = i % C3, hd = f >> 6;
    gout[i] = gout[i] / denom[node * HEADS + hd] + bias[f];
  } }

// ===========================================================================
extern "C" void kernel_launch(void* const* d_in, const int* in_sizes, int n_in,
                              void* d_out, int out_size, void* d_ws, size_t ws_size,
                              hipStream_t stream) {
  const float* x     = (const float*)d_in[0];
  const int*   src   = (const int*)  d_in[1];
  const int*   dst   = (const int*)  d_in[2];
  const float* embW  = (const float*)d_in[3];
  const float* embB  = (const float*)d_in[4];
  const float* chebW = (const float*)d_in[5];
  const float* chebB = (const float*)d_in[6];
  const float* glw   = (const float*)d_in[7];
  const float* gas   = (const float*)d_in[8];
  const float* gad   = (const float*)d_in[9];
  const float* gbias = (const float*)d_in[10];
  const float* sbg   = (const float*)d_in[11];
  const float* sbb   = (const float*)d_in[12];
  const float* slw   = (const float*)d_in[13];
  const float* slb   = (const float*)d_in[14];
  const float* bng   = (const float*)d_in[15];
  const float* bnb   = (const float*)d_in[16];
  float* out = (float*)d_out;

  // ---------------- workspace carve-up (floats) ----------------
  float* ws = (float*)d_ws;
  size_t off = 0;
  auto alloc = [&](size_t n) { float* p = ws + off; off += n; return p; };
  const size_t NH = (size_t)NNODE * H;
  float* t1b  = alloc(NH);
  float* t2b  = alloc(NH);
  float* tb   = alloc(NH);   // cheb/BN output t_i
  float* ybuf = alloc(NH);   // y (embedding)
  float* yF   = alloc(NH);
  float* y2F  = alloc(NH);   // later reused as y8
  float* y4F  = alloc(NH);
  float* y6F  = alloc(NH);
  float* ytmp = alloc(NH);
  float* ygat = alloc(NH);
  float* hbuf = alloc((size_t)NNODE * C3);
  float* gout = alloc((size_t)NNODE * C3);
  float* asrc = alloc((size_t)NNODE * HEADS);
  float* adst = alloc((size_t)NNODE * HEADS);
  float* mbuf = alloc((size_t)NNODE * HEADS);
  float* dnm  = alloc((size_t)NNODE * HEADS);
  float* pbuf = alloc((size_t)(NE + NNODE) * HEADS);
  float* disb = alloc(NNODE);
  float* wbuf = alloc(NE);
  float* colS = alloc(C3);
  float* colQ = alloc(C3);
  (void)ws_size; (void)in_sizes; (void)n_in; (void)out_size;

  const int TB = 256;
  auto nb = [](long long n) { return (unsigned)((n + 255) / 256); };

  // ---------------- helpers ----------------
  auto bn = [&](float* X, const float* g, const float* b, int C) {
    hipMemsetAsync(colS, 0, C * sizeof(float), stream);
    hipMemsetAsync(colQ, 0, C * sizeof(float), stream);
    const int rpb = 128;
    k_colstats<<<(NNODE + rpb - 1) / rpb, C, 0, stream>>>(X, colS, colQ, NNODE, C, rpb);
    k_bnapply<<<nb((long long)NNODE * C), TB, 0, stream>>>(X, colS, colQ, g, b, NNODE, C);
  };

  auto cheb = [&](const float* zin, int j, float* tout) {
    hipMemsetAsync(t1b, 0, NH * sizeof(float), stream);
    k_lap<<<nb((long long)NE * H), TB, 0, stream>>>(src, dst, wbuf, zin, t1b, 1.0f, NE);
    k_neg<<<nb(NH), TB, 0, stream>>>(zin, t2b, (int)NH);        // t2 = -z ...
    k_lap<<<nb((long long)NE * H), TB, 0, stream>>>(src, dst, wbuf, t1b, t2b, 2.0f, NE); // ... + 2*lap(t1)
    const float* W = chebW + (size_t)j * 3 * H * H;
    gemm_wmma<4, 0><<<NNODE / 16, 32, 0, stream>>>(zin, W,             chebB + j * H, tout, H, 0);
    gemm_wmma<4, 0><<<NNODE / 16, 32, 0, stream>>>(t1b, W +     H * H, nullptr,       tout, H, 2);
    gemm_wmma<4, 0><<<NNODE / 16, 32, 0, stream>>>(t2b, W + 2 * H * H, nullptr,       tout, H, 2);
    bn(tout, bng, bnb, H);
  };

  auto gatseq = [&](const float* zin, int i) {
    // h = z @ Wl^T : [N,64] x [192,64]^T -> [N,192]
    gemm_wmma<12, 1><<<NNODE / 16, 32, 0, stream>>>(zin, glw + (size_t)i * C3 * H, nullptr, hbuf, H, 0);
    k_att<<<nb((long long)NNODE * HEADS), TB, 0, stream>>>(hbuf, gas + (size_t)i * C3, gad + (size_t)i * C3,
                                                           asrc, adst, NNODE);
    k_fill<<<nb((long long)NNODE * HEADS), TB, 0, stream>>>(mbuf, -__builtin_huge_valf(), NNODE * HEADS);
    hipMemsetAsync(dnm, 0, (size_t)NNODE * HEADS * sizeof(float), stream);
    // softmax pass 1: segment max (real edges, then self loops)
    k_gmax<<<nb((long long)NE * HEADS), TB, 0, stream>>>(src, dst, asrc, adst, mbuf, NE);
    k_gmax<<<nb((long long)NNODE * HEADS), TB, 0, stream>>>(nullptr, nullptr, asrc, adst, mbuf, NNODE);
    // pass 2: p = exp(e - m), denom = segment sum
    k_gexp<<<nb((long long)NE * HEADS), TB, 0, stream>>>(src, dst, asrc, adst, mbuf, pbuf, dnm, NE);
    k_gexp<<<nb((long long)NNODE * HEADS), TB, 0, stream>>>(nullptr, nullptr, asrc, adst, mbuf,
                                                            pbuf + (size_t)NE * HEADS, dnm, NNODE);
    // pass 3: numerator scatter
    hipMemsetAsync(gout, 0, (size_t)NNODE * C3 * sizeof(float), stream);
    k_gnum<<<nb((long long)NE * C3), TB, 0, stream>>>(src, dst, pbuf, hbuf, gout, NE);
    k_gnum<<<nb((long long)NNODE * C3), TB, 0, stream>>>(nullptr, nullptr, pbuf + (size_t)NE * HEADS,
                                                         hbuf, gout, NNODE);
    k_gfin<<<nb((long long)NNODE * C3), TB, 0, stream>>>(gout, dnm, gbias + (size_t)i * C3, NNODE);
    bn(gout, sbg + (size_t)i * C3, sbb + (size_t)i * C3, C3);
    // ygat += gout @ seq_lin_w^T + seq_lin_b
    gemm_wmma<4, 1><<<NNODE / 16, 32, 0, stream>>>(gout, slw + (size_t)i * H * C3,
                                                   slb + (size_t)i * H, ygat, C3, 2);
  };

  // ---------------- edge weights (once) ----------------
  hipMemsetAsync(disb, 0, NNODE * sizeof(float), stream);
  k_deg<<<nb(NE), TB, 0, stream>>>(src, disb, NE);
  k_dis<<<nb(NNODE), TB, 0, stream>>>(disb, NNODE);
  k_ew<<<nb(NE), TB, 0, stream>>>(src, dst, disb, wbuf, NE);
  hipMemsetAsync(ygat, 0, NH * sizeof(float), stream);

  // ---------------- forward ----------------
  // y = relu(x @ emb_w^T + emb_b), K=200
  gemm_wmma<4, 1><<<NNODE / 16, 32, 0, stream>>>(x, embW, embB, ybuf, FIN, 1);

  const int n = (int)NH;
  cheb(ybuf, 0, tb); gatseq(tb, 0); k_relu<<<nb(n), TB, 0, stream>>>(tb, yF, n);
  cheb(yF,   1, tb); gatseq(tb, 1); k_relu<<<nb(n), TB, 0, stream>>>(tb, ytmp, n);      // y1
  cheb(ytmp, 2, tb); gatseq(yF, 2);                                                     // seq2 takes yF (source bug)
  k_addrelu<<<nb(n), TB, 0, stream>>>(tb, yF, y2F, n);                                  // y2F = relu(t2+yF)
  cheb(y2F,  3, tb); gatseq(tb, 3); k_relu<<<nb(n), TB, 0, stream>>>(tb, ytmp, n);      // y3
  cheb(ytmp, 4, tb); gatseq(tb, 4); k_addrelu<<<nb(n), TB, 0, stream>>>(y2F, tb, y4F, n);
  cheb(y4F,  5, tb); gatseq(tb, 5); k_relu<<<nb(n), TB, 0, stream>>>(tb, ytmp, n);      // y5
  cheb(ytmp, 6, tb); gatseq(tb, 6); k_reluadd<<<nb(n), TB, 0, stream>>>(tb, y4F, y6F, n);
  cheb(y6F,  7, tb); gatseq(tb, 7); k_relu<<<nb(n), TB, 0, stream>>>(tb, ytmp, n);      // y7
  cheb(ytmp, 7, tb); gatseq(tb, 8);                                                     // block 8 reuses conv7
  k_reluadd<<<nb(n), TB, 0, stream>>>(tb, y6F, y2F, n);                                 // y8 (reuse y2F)

  k_mulrelu<<<nb(n), TB, 0, stream>>>(y2F, ygat, out, n);                               // out = y8 * relu(ygat)
}